// FuzzyClustering_62646392979771
// MI455X (gfx1250) — compile-verified
//
#include <hip/hip_runtime.h>
#include <hip/hip_bf16.h>
#include <math.h>

typedef float v2f __attribute__((ext_vector_type(2)));
typedef float v8f __attribute__((ext_vector_type(8)));

constexpr int kB = 8, kM = 64, kD = 256, kN = 16384;
constexpr int kPT = 36;   // ptile row stride (floats): 144 B = 9*16 B -> B128-aligned rows

#define WMMA_F32(a, b, c) \
    __builtin_amdgcn_wmma_f32_16x16x4_f32(false, (a), false, (b), (short)0, (c), false, false)

// ---------------------------------------------------------------------------
// K0: zero P and ssum accumulators (they are contiguous in the workspace)
// ---------------------------------------------------------------------------
__global__ void zero_kernel(float* __restrict__ p, int n) {
    int i = blockIdx.x * 256 + threadIdx.x;
    if (i < n) p[i] = 0.f;
}

// ---------------------------------------------------------------------------
// K1: c = SiLU((clusters+cbias)@Wc^T + bc); cn = alpha * l2norm(c)
//     vc = SiLU((vclusters+vbias)@Wv^T + bv)
// one block per (b,m) row, 256 threads (one per output d)
// ---------------------------------------------------------------------------
__global__ __launch_bounds__(256) void cluster_proj_kernel(
    const float* __restrict__ clusters, const float* __restrict__ cbias,
    const float* __restrict__ Wc, const float* __restrict__ bc,
    const float* __restrict__ vclusters, const float* __restrict__ vbias,
    const float* __restrict__ Wv, const float* __restrict__ bv,
    const float* __restrict__ alpha,
    float* __restrict__ cn, float* __restrict__ vc)
{
    int b = blockIdx.x / kM, m = blockIdx.x % kM, d = threadIdx.x;
    __shared__ float xin[kD];
    __shared__ float red[kD];

    // ---- c path ----
    xin[d] = clusters[(b * kM + m) * kD + d] + cbias[m * kD + d];
    __syncthreads();
    float acc = bc[d];
    const float* wrow = Wc + d * kD;
    for (int k = 0; k < kD; ++k) acc = fmaf(xin[k], wrow[k], acc);
    float y = acc / (1.f + __expf(-acc));           // SiLU
    red[d] = y * y;
    __syncthreads();
    for (int s = 128; s > 0; s >>= 1) {
        if (d < s) red[d] += red[d + s];
        __syncthreads();
    }
    float scale = alpha[m] / fmaxf(sqrtf(red[0]), 1e-12f);
    cn[(b * kM + m) * kD + d] = y * scale;
    __syncthreads();

    // ---- vc path ----
    xin[d] = vclusters[(b * kM + m) * kD + d] + vbias[m * kD + d];
    __syncthreads();
    float acc2 = bv[d];
    const float* wrow2 = Wv + d * kD;
    for (int k = 0; k < kD; ++k) acc2 = fmaf(xin[k], wrow2[k], acc2);
    vc[(b * kM + m) * kD + d] = acc2 / (1.f + __expf(-acc2));
}

// ---------------------------------------------------------------------------
// K2: sim = cn @ l2norm(points, over D) + beta  -> softmax over M
//     write assignment [B,M,N] to ws; atomic per-(b,m) row sums
// grid (N/32, B), 256 threads = 8 waves: 4 M-tiles x 2 N-tiles of 16x16
// Points tile staged with async global->LDS (ASYNCcnt path, no VGPR bounce).
// ---------------------------------------------------------------------------
__global__ __launch_bounds__(256) void sim_softmax_kernel(
    const float* __restrict__ points, const float* __restrict__ cn,
    const float* __restrict__ sbeta,
    float* __restrict__ Aout, float* __restrict__ ssum)
{
    __shared__ float ptile[kD][kPT];  // points tile, rows 16B-aligned for B128
    __shared__ float rinv[32];        // per-column 1/||p||
    __shared__ float simbuf[kM][33];  // 64 x 32 sim tile

    int b = blockIdx.y;
    int n0 = blockIdx.x * 32;
    int t = threadIdx.x;

    // stage points tile [256 d][32 n] via async global->LDS, 16B per lane
    {
        int q = t & 7;          // 16B chunk within row (4 floats)
        int dr = t >> 3;        // 32 rows per pass
        for (int d = dr; d < kD; d += 32) {
            unsigned lds = (unsigned)(uintptr_t)&ptile[d][q * 4];
            const float* g = points + (size_t)(b * kD + d) * kN + n0 + q * 4;
            asm volatile("global_load_async_to_lds_b128 %0, %1, off"
                         :: "v"(lds), "v"(g) : "memory");
        }
        asm volatile("s_wait_asynccnt 0x0" ::: "memory");
    }
    __syncthreads();

    if (t < 32) {
        float s = 0.f;
        for (int d = 0; d < kD; ++d) { float v = ptile[d][t]; s = fmaf(v, v, s); }
        rinv[t] = 1.f / fmaxf(sqrtf(s), 1e-12f);
    }
    __syncthreads();

    int wid = t >> 5, lane = t & 31;
    int lhalf = lane >> 4, l16 = lane & 15;
    int m0 = (wid & 3) * 16;        // M-tile within 64
    int nb = (wid >> 2) * 16;       // N-tile within 32

    const float* cnb = cn + (size_t)b * kM * kD;
    v8f acc = {0.f, 0.f, 0.f, 0.f, 0.f, 0.f, 0.f, 0.f};
    for (int k = 0; k < kD; k += 4) {
        int ka = k + 2 * lhalf;
        v2f a = *(const v2f*)(cnb + (m0 + l16) * kD + ka);   // A[m][k], K pair
        float rv = rinv[nb + l16];
        v2f bb;
        bb.x = ptile[ka][nb + l16] * rv;                      // B[k][n]
        bb.y = ptile[ka + 1][nb + l16] * rv;
        acc = WMMA_F32(a, bb, acc);
    }

    // scatter 16x16 f32 C tile into simbuf, add beta
    for (int r = 0; r < 8; ++r) {
        int ml = m0 + r + 8 * lhalf;
        simbuf[ml][nb + l16] = acc[r] + sbeta[ml];
    }
    __syncthreads();

    // softmax over M (64) per column, serial (M is tiny)
    if (t < 32) {
        float mx = -3.4e38f;
        for (int m = 0; m < kM; ++m) mx = fmaxf(mx, simbuf[m][t]);
        float s = 0.f;
        for (int m = 0; m < kM; ++m) {
            float e = __expf(simbuf[m][t] - mx);
            simbuf[m][t] = e;
            s += e;
        }
        float inv = 1.f / s;
        for (int m = 0; m < kM; ++m) simbuf[m][t] *= inv;
    }
    __syncthreads();

    // per-m partial row sums -> global atomic
    if (t < kM) {
        float s = 0.f;
        for (int n = 0; n < 32; ++n) s += simbuf[t][n];
        atomicAdd(&ssum[b * kM + t], s);
    }

    // write assignment tile, coalesced per m-row
    {
        int n = t & 31, mr = t >> 5;
        for (int m = mr; m < kM; m += 8)
            Aout[(size_t)(b * kM + m) * kN + n0 + n] = simbuf[m][n];
    }
}

// ---------------------------------------------------------------------------
// K3: P += assignment @ values  ([64,N] x [N,256] -> [64,256] per b)
// grid (32 N-chunks, B); each block does full [64,256] partial over 512 n's
// ---------------------------------------------------------------------------
__global__ __launch_bounds__(256) void av_kernel(
    const float* __restrict__ Aw, const float* __restrict__ values,
    float* __restrict__ P)
{
    int b = blockIdx.y;
    int nbase = blockIdx.x * (kN / 32);   // 512-wide chunk
    int t = threadIdx.x, wid = t >> 5, lane = t & 31;
    int lhalf = lane >> 4, l16 = lane & 15;
    const float* Ab = Aw + (size_t)b * kM * kN;
    const float* Vb = values + (size_t)b * kN * kD;

    for (int ti = wid; ti < 64; ti += 8) {      // 4 M-tiles x 16 D-tiles
        int m0 = (ti >> 4) * 16;
        int d0 = (ti & 15) * 16;
        v8f acc = {0.f, 0.f, 0.f, 0.f, 0.f, 0.f, 0.f, 0.f};
        for (int k = nbase; k < nbase + kN / 32; k += 4) {
            int ka = k + 2 * lhalf;
            v2f a = *(const v2f*)(Ab + (size_t)(m0 + l16) * kN + ka);  // A[m][n]
            v2f bb;
            bb.x = Vb[(size_t)ka * kD + d0 + l16];                     // B[n][d]
            bb.y = Vb[(size_t)(ka + 1) * kD + d0 + l16];
            acc = WMMA_F32(a, bb, acc);
        }
        for (int r = 0; r < 8; ++r)
            atomicAdd(&P[((size_t)b * kM + m0 + r + 8 * lhalf) * kD + d0 + l16], acc[r]);
    }
}

// ---------------------------------------------------------------------------
// K4: agg = (vc + P) / (ssum + eps)
// ---------------------------------------------------------------------------
__global__ void agg_kernel(const float* __restrict__ vc, const float* __restrict__ P,
                           const float* __restrict__ ssum, float* __restrict__ agg)
{
    int i = blockIdx.x * 256 + threadIdx.x;  // over B*M*D
    int bm = i / kD;
    agg[i] = (vc[i] + P[i]) / (ssum[bm] + 1e-7f);
}

// ---------------------------------------------------------------------------
// K5: x[b,d,n] = sum_m agg[b,m,d] * assignment[b,m,n]
// grid (N/32, D/64, B); block = 8 waves = 4 D-tiles x 2 N-tiles; K = M = 64
// ---------------------------------------------------------------------------
__global__ __launch_bounds__(256) void out_kernel(
    const float* __restrict__ agg, const float* __restrict__ Aw,
    float* __restrict__ xout)
{
    int b = blockIdx.z;
    int d0 = blockIdx.y * 64;
    int n0 = blockIdx.x * 32;
    int t = threadIdx.x, wid = t >> 5, lane = t & 31;
    int lhalf = lane >> 4, l16 = lane & 15;
    int dw = d0 + (wid & 3) * 16;
    int nw = n0 + (wid >> 2) * 16;
    const float* gb = agg + (size_t)b * kM * kD;
    const float* Ab = Aw + (size_t)b * kM * kN;

    v8f acc = {0.f, 0.f, 0.f, 0.f, 0.f, 0.f, 0.f, 0.f};
    for (int k = 0; k < kM; k += 4) {
        int ka = k + 2 * lhalf;
        v2f a, bb;
        a.x = gb[ka * kD + dw + l16];            // A[d][m] = agg[m][d] (transposed)
        a.y = gb[(ka + 1) * kD + dw + l16];
        bb.x = Ab[(size_t)ka * kN + nw + l16];   // B[m][n]
        bb.y = Ab[(size_t)(ka + 1) * kN + nw + l16];
        acc = WMMA_F32(a, bb, acc);
    }
    for (int r = 0; r < 8; ++r)
        xout[((size_t)b * kD + dw + r + 8 * lhalf) * kN + nw + l16] = acc[r];
}

// ---------------------------------------------------------------------------
extern "C" void kernel_launch(void* const* d_in, const int* in_sizes, int n_in,
                              void* d_out, int out_size, void* d_ws, size_t ws_size,
                              hipStream_t stream) {
    const float* points    = (const float*)d_in[0];   // [B,D,N]
    const float* clusters  = (const float*)d_in[1];   // [B,M,D]
    const float* values    = (const float*)d_in[2];   // [B,N,D]
    const float* vclusters = (const float*)d_in[3];   // [B,M,D]
    const float* alpha     = (const float*)d_in[4];   // [M]
    const float* beta      = (const float*)d_in[5];   // [M]
    const float* cbias     = (const float*)d_in[6];   // [M,D]
    const float* Wc        = (const float*)d_in[7];   // [D,D]
    const float* bc        = (const float*)d_in[8];   // [D]
    const float* vbias     = (const float*)d_in[9];   // [M,D]
    const float* Wv        = (const float*)d_in[10];  // [D,D]
    const float* bv        = (const float*)d_in[11];  // [D]
    float* xout = (float*)d_out;

    // workspace layout (floats)
    float* ws   = (float*)d_ws;
    float* cn   = ws;                                   // B*M*D
    float* vc   = cn + (size_t)kB * kM * kD;            // B*M*D
    float* Aws  = vc + (size_t)kB * kM * kD;            // B*M*N
    float* P    = Aws + (size_t)kB * kM * kN;           // B*M*D
    float* ssum = P + (size_t)kB * kM * kD;             // B*M
    float* agg  = ssum + (size_t)kB * kM;               // B*M*D

    // zero P + ssum (contiguous)
    int nz = kB * kM * kD + kB * kM;
    zero_kernel<<<(nz + 255) / 256, 256, 0, stream>>>(P, nz);

    cluster_proj_kernel<<<kB * kM, 256, 0, stream>>>(
        clusters, cbias, Wc, bc, vclusters, vbias, Wv, bv, alpha, cn, vc);

    sim_softmax_kernel<<<dim3(kN / 32, kB), 256, 0, stream>>>(
        points, cn, beta, Aws, ssum);

    av_kernel<<<dim3(32, kB), 256, 0, stream>>>(Aws, values, P);

    agg_kernel<<<(kB * kM * kD) / 256, 256, 0, stream>>>(vc, P, ssum, agg);

    out_kernel<<<dim3(kN / 32, kD / 64, kB), 256, 0, stream>>>(agg, Aws, xout);
}